// GNN_13048110645409
// MI455X (gfx1250) — compile-verified
//
#include <hip/hip_runtime.h>

typedef __attribute__((ext_vector_type(2))) float v2f;
typedef __attribute__((ext_vector_type(8))) float v8f;

#define DIM_IN  256
#define DIM_HID 128
#define DIM_CLS 40

// ---------------------------------------------------------------- utilities

__global__ void zero_kernel(float* __restrict__ p, long n) {
    long i = (long)blockIdx.x * blockDim.x + threadIdx.x;
    if (i < n) p[i] = 0.0f;
}

__global__ void count_deg_kernel(const long long* __restrict__ dst,
                                 int* __restrict__ cnt, int E) {
    int e = blockIdx.x * blockDim.x + threadIdx.x;
    if (e < E) atomicAdd(&cnt[(int)dst[e]], 1);
}

__global__ void dinv_kernel(const int* __restrict__ cnt,
                            float* __restrict__ dinv, int N) {
    int i = blockIdx.x * blockDim.x + threadIdx.x;
    if (i < N) dinv[i] = rsqrtf((float)(cnt[i] + 1));  // +1 self-loop; always > 0
}

// ---------------------------------------------------------------- WMMA GEMM
// C[M,NVALID] = A[M,KDIM] x B[KDIM,NVALID], fp32 via V_WMMA_F32_16X16X4_F32.
// Block: (NCOLS/16) waves; each wave owns one 16x16 output tile.
// A tile (16 rows x KDIM) staged in LDS with stride KDIM+2 (bank-conflict pad).
template <int KDIM, int NCOLS, int NVALID>
__global__ void gemm_wmma_kernel(const float* __restrict__ A,
                                 const float* __restrict__ B,
                                 float* __restrict__ C, int M) {
    constexpr int LDSS = KDIM + 2;
    __shared__ float ldsA[16 * LDSS];

    const int tid     = threadIdx.x;
    const int rowBase = blockIdx.x * 16;

    // Cooperative load of 16 x KDIM A tile (float4 global reads).
    for (int i = tid; i < 16 * (KDIM / 4); i += blockDim.x) {
        int r  = i / (KDIM / 4);
        int c4 = (i % (KDIM / 4)) * 4;
        int rr = rowBase + r; if (rr > M - 1) rr = M - 1;   // clamp (tail safety)
        const float4 v = *reinterpret_cast<const float4*>(A + (size_t)rr * KDIM + c4);
        float* p = &ldsA[r * LDSS + c4];
        p[0] = v.x; p[1] = v.y; p[2] = v.z; p[3] = v.w;
    }
    __syncthreads();

    const int wave = tid >> 5;
    const int lane = tid & 31;
    const int half = lane >> 4;     // 0: K=0..1 / M=r ; 1: K=2..3 / M=r+8
    const int l16  = lane & 15;
    const int col  = wave * 16 + l16;
    const bool cvalid = (NVALID == NCOLS) || (col < NVALID);

    v8f acc = {0.f, 0.f, 0.f, 0.f, 0.f, 0.f, 0.f, 0.f};
    const float* arow = &ldsA[l16 * LDSS];

#pragma unroll 8
    for (int k0 = 0; k0 < KDIM; k0 += 4) {
        const int kb = k0 + 2 * half;
        v2f a, b;
        a.x = arow[kb];
        a.y = arow[kb + 1];
        if (cvalid) {
            b.x = B[(size_t)kb * NVALID + col];
            b.y = B[(size_t)(kb + 1) * NVALID + col];
        } else {
            b.x = 0.f; b.y = 0.f;
        }
        // 8 args: (neg_a, A, neg_b, B, c_mod, C, reuse_a, reuse_b)
        acc = __builtin_amdgcn_wmma_f32_16x16x4_f32(
            false, a, false, b, (short)0, acc, false, false);
    }

    // C/D layout: VGPR r holds (M = r + 8*half, N = l16).
    // Fast path (the only one hit for M % 16 == 0): 8 unguarded stores.
    if (cvalid) {
        const int rowTop = rowBase + half * 8;
        float* Cp = C + (size_t)rowTop * NVALID + col;
        if (rowBase + 16 <= M) {
#pragma unroll
            for (int r = 0; r < 8; ++r)
                Cp[(size_t)r * NVALID] = acc[r];
        } else {
#pragma unroll
            for (int r = 0; r < 8; ++r)
                if (rowTop + r < M) Cp[(size_t)r * NVALID] = acc[r];
        }
    }
}

// ------------------------------------------------------- edge scatter-add
// One wave per edge: float4 gather of h[src]*w, scalar f32 atomics into out[dst].
__global__ void aggregate_kernel(const float* __restrict__ h,
                                 float* __restrict__ out,
                                 const long long* __restrict__ src,
                                 const long long* __restrict__ dst,
                                 const float* __restrict__ dinv,
                                 int E, int D) {
    int e    = blockIdx.x * (blockDim.x >> 5) + (threadIdx.x >> 5);
    int lane = threadIdx.x & 31;
    if (e >= E) return;
    int s = (int)src[e];
    int d = (int)dst[e];
    float w = dinv[s] * dinv[d];
    const float* hs = h + (size_t)s * D;
    float* od = out + (size_t)d * D;
    for (int c = lane * 4; c < D; c += 128) {
        float4 v = *reinterpret_cast<const float4*>(hs + c);
        atomicAdd(&od[c + 0], v.x * w);
        atomicAdd(&od[c + 1], v.y * w);
        atomicAdd(&od[c + 2], v.z * w);
        atomicAdd(&od[c + 3], v.w * w);
    }
}

// ------------------------------------------- accumulator init: self-loop + bias
// out[n,c] = h[n,c] * dinv[n]^2 + bias[c]   (edge sum is added on top afterwards)
__global__ void init_selfloop_kernel(float* __restrict__ out,
                                     const float* __restrict__ h,
                                     const float* __restrict__ dinv,
                                     const float* __restrict__ bias,
                                     long total, int D) {
    long i = (long)blockIdx.x * blockDim.x + threadIdx.x;
    if (i >= total) return;
    int n = (int)(i / D);
    int c = (int)(i % D);
    float di = dinv[n];
    out[i] = fmaf(h[i], di * di, bias[c]);
}

__global__ void relu_kernel(float* __restrict__ p, long n) {
    long i = (long)blockIdx.x * blockDim.x + threadIdx.x;
    if (i < n) p[i] = fmaxf(p[i], 0.0f);
}

// ---------------------------------------------------------------- driver

extern "C" void kernel_launch(void* const* d_in, const int* in_sizes, int n_in,
                              void* d_out, int out_size, void* d_ws, size_t ws_size,
                              hipStream_t stream) {
    const float*     x   = (const float*)d_in[0];
    const long long* ei  = (const long long*)d_in[1];   // jnp.int64 edge_index [2,E]
    const float*     W1  = (const float*)d_in[2];
    const float*     b1  = (const float*)d_in[3];
    const float*     W2  = (const float*)d_in[4];
    const float*     b2  = (const float*)d_in[5];
    float*           out = (float*)d_out;

    const int N = in_sizes[0] / DIM_IN;   // 100000
    const int E = in_sizes[1] / 2;        // 1600000
    const long long* srcE = ei;
    const long long* dstE = ei + E;

    // Workspace carve-up (~103 MB): dinv | cnt | h1 (reused as h2) | h1agg
    char* ws = (char*)d_ws;
    float* dinv  = (float*)ws;  ws += (size_t)N * sizeof(float);
    int*   cnt   = (int*)ws;    ws += (size_t)N * sizeof(int);
    float* h1    = (float*)ws;  ws += (size_t)N * DIM_HID * sizeof(float);
    float* h1agg = (float*)ws;  ws += (size_t)N * DIM_HID * sizeof(float);
    float* h2 = h1;             // h1 is dead after aggregate(layer1); reuse for layer-2

    const int TB = 256;
    const int rowTiles = (N + 15) / 16;
    const int edgeWaveBlocks = (E + 7) / 8;   // 8 waves/block, 1 wave per edge

    // degree (with self-loop folded in as +1) -> dinv
    zero_kernel<<<(N + TB - 1) / TB, TB, 0, stream>>>((float*)cnt, (long)N);
    count_deg_kernel<<<(E + TB - 1) / TB, TB, 0, stream>>>(dstE, cnt, E);
    dinv_kernel<<<(N + TB - 1) / TB, TB, 0, stream>>>(cnt, dinv, N);

    // layer 1: h1 = x @ W1 (WMMA fp32); agg = selfloop + b1 + edge sum; ReLU
    gemm_wmma_kernel<DIM_IN, DIM_HID, DIM_HID>
        <<<rowTiles, (DIM_HID / 16) * 32, 0, stream>>>(x, W1, h1, N);
    init_selfloop_kernel<<<((long)N * DIM_HID + TB - 1) / TB, TB, 0, stream>>>(
        h1agg, h1, dinv, b1, (long)N * DIM_HID, DIM_HID);
    aggregate_kernel<<<edgeWaveBlocks, TB, 0, stream>>>(h1, h1agg, srcE, dstE, dinv, E, DIM_HID);
    relu_kernel<<<((long)N * DIM_HID + TB - 1) / TB, TB, 0, stream>>>(
        h1agg, (long)N * DIM_HID);

    // layer 2: h2 = h1agg @ W2 (N=40 padded to 48 cols); out = selfloop + b2 + edge sum
    gemm_wmma_kernel<DIM_HID, 48, DIM_CLS>
        <<<rowTiles, (48 / 16) * 32, 0, stream>>>(h1agg, W2, h2, N);
    init_selfloop_kernel<<<((long)N * DIM_CLS + TB - 1) / TB, TB, 0, stream>>>(
        out, h2, dinv, b2, (long)N * DIM_CLS, DIM_CLS);
    aggregate_kernel<<<edgeWaveBlocks, TB, 0, stream>>>(h2, out, srcE, dstE, dinv, E, DIM_CLS);
}